// STKGRec_33165737460368
// MI455X (gfx1250) — compile-verified
//
#include <hip/hip_runtime.h>

typedef __attribute__((ext_vector_type(2))) float v2f;
typedef __attribute__((ext_vector_type(8))) float v8f;

#define S_DIM 1024
#define H_DIM 128
#define LDS_STRIDE 132   // 128 + 4 pad -> conflict-free ds_load_b64 column reads
#define N_BLOCKS 16      // one block per 16-row batch tile

__global__ void gru_bar_init(unsigned int* bar) {
  if (threadIdx.x == 0) *bar = 0u;
}

__global__ __launch_bounds__(256, 1)
void dilated_gru_wmma(const float* __restrict__ x,
                      const float* __restrict__ w_ih,
                      const float* __restrict__ w_hh,
                      const float* __restrict__ b_ih,
                      const float* __restrict__ b_hh,
                      const int*   __restrict__ idx,
                      float*       __restrict__ out,
                      unsigned int* __restrict__ bar)
{
  __shared__ __align__(16) float sx[16][LDS_STRIDE];  // x_t tile  [16 batch rows][128]
  __shared__ __align__(16) float sh[16][LDS_STRIDE];  // gathered h_prev tile

  const int bm   = blockIdx.x;        // batch tile: rows bm*16 .. bm*16+15
  const int tid  = threadIdx.x;
  const int wave = tid >> 5;          // 0..7 : H column tile (16 cols each)
  const int lane = tid & 31;
  const int hl   = lane >> 4;         // lane half (K-pair select per ISA 16x4 A layout)
  const int ln   = lane & 15;

  // staging map: 16 threads per row, 8 floats per thread
  const int sr = tid >> 4;            // 0..15
  const int sc = (tid & 15) << 3;     // 0,8,...,120

  // N column this lane produces: B[k][n] = W[n_global][k]
  const int ncol = (wave << 4) + ln;  // 0..127
  const float* wr_i = w_ih + (size_t)(ncol            ) * H_DIM;
  const float* wz_i = w_ih + (size_t)(ncol +   H_DIM  ) * H_DIM;
  const float* wn_i = w_ih + (size_t)(ncol + 2*H_DIM  ) * H_DIM;
  const float* wr_h = w_hh + (size_t)(ncol            ) * H_DIM;
  const float* wz_h = w_hh + (size_t)(ncol +   H_DIM  ) * H_DIM;
  const float* wn_h = w_hh + (size_t)(ncol + 2*H_DIM  ) * H_DIM;

  const float bias_r   = b_ih[ncol]           + b_hh[ncol];
  const float bias_z   = b_ih[ncol + H_DIM]   + b_hh[ncol + H_DIM];
  const float bias_n_i = b_ih[ncol + 2*H_DIM];
  const float bias_n_h = b_hh[ncol + 2*H_DIM];

  const int srow_b = bm * 16 + sr;                        // global batch row (staging)
  const float* x_stage = x + (size_t)srow_b * S_DIM * H_DIM + sc;
  const int*   idx_row = idx + (size_t)srow_b * S_DIM;

  const int kofs = hl << 1;           // A/B fragment K = {2*hl, 2*hl+1}
  const v8f vzero = {};

  // ---- prologue: stage x_0, compute gi_0 (no recurrence dependency) ----
  {
    const float* xp = x_stage;        // t = 0
    *(float4*)&sx[sr][sc]     = ((const float4*)xp)[0];
    *(float4*)&sx[sr][sc + 4] = ((const float4*)xp)[1];
    __builtin_prefetch(xp + H_DIM, 0, 0);
  }
  __syncthreads();

  v8f a_ir = vzero, a_iz = vzero, a_in = vzero;
  #pragma unroll 4
  for (int kk = 0; kk < H_DIM; kk += 4) {
    const int kb = kk + kofs;
    v2f ax  = *(const v2f*)&sx[ln][kb];
    v2f bri = *(const v2f*)(wr_i + kb);
    v2f bzi = *(const v2f*)(wz_i + kb);
    v2f bni = *(const v2f*)(wn_i + kb);
    a_ir = __builtin_amdgcn_wmma_f32_16x16x4_f32(false, ax, false, bri, (short)0, a_ir, false, false);
    a_iz = __builtin_amdgcn_wmma_f32_16x16x4_f32(false, ax, false, bzi, (short)0, a_iz, false, false);
    a_in = __builtin_amdgcn_wmma_f32_16x16x4_f32(false, ax, false, bni, (short)0, a_in, false, false);
  }

  for (int t = 0; t < S_DIM; ++t) {
    // ---------- stage gathered h_prev (serial-critical) ----------
    if (t == 0) {
      const float4 z4 = {0.f, 0.f, 0.f, 0.f};
      *(float4*)&sh[sr][sc]     = z4;                      // reference: zero init state
      *(float4*)&sh[sr][sc + 4] = z4;
    } else {
      const int iv = idx_row[t];                           // guaranteed < t
      const float* hp = out + ((size_t)srow_b * S_DIM + iv) * H_DIM + sc;
      *(float4*)&sh[sr][sc]     = ((const float4*)hp)[0];  // L2-resident gather
      *(float4*)&sh[sr][sc + 4] = ((const float4*)hp)[1];
    }
    __syncthreads();

    // ---------- gh GEMM: the only WMMA work on the serial path ----------
    v8f a_hr = vzero, a_hz = vzero, a_hn = vzero;
    #pragma unroll 4
    for (int kk = 0; kk < H_DIM; kk += 4) {
      const int kb = kk + kofs;
      v2f ah  = *(const v2f*)&sh[ln][kb];
      v2f brh = *(const v2f*)(wr_h + kb);
      v2f bzh = *(const v2f*)(wz_h + kb);
      v2f bnh = *(const v2f*)(wn_h + kb);
      a_hr = __builtin_amdgcn_wmma_f32_16x16x4_f32(false, ah, false, brh, (short)0, a_hr, false, false);
      a_hz = __builtin_amdgcn_wmma_f32_16x16x4_f32(false, ah, false, bzh, (short)0, a_hz, false, false);
      a_hn = __builtin_amdgcn_wmma_f32_16x16x4_f32(false, ah, false, bnh, (short)0, a_hn, false, false);
    }

    // ---------- fused GRU gate epilogue + store h_new ----------
    #pragma unroll
    for (int v = 0; v < 8; ++v) {
      const int m = v + (hl << 3);             // C/D row per ISA 16x16 f32 layout
      const float hprev = sh[m][ncol];
      const float rs = a_ir[v] + a_hr[v] + bias_r;
      const float zs = a_iz[v] + a_hz[v] + bias_z;
      const float rg = 1.0f / (1.0f + __expf(-rs));        // sigmoid (saturation-safe)
      const float zg = 1.0f / (1.0f + __expf(-zs));
      const float nn = (a_in[v] + bias_n_i) + rg * (a_hn[v] + bias_n_h);
      const float e2 = __expf(-2.0f * nn);
      const float nc = 2.0f / (1.0f + e2) - 1.0f;          // tanh, safe at +/-inf
      const float hnew = (1.0f - zg) * nc + zg * hprev;
      out[(((size_t)(bm * 16 + m)) * S_DIM + t) * H_DIM + ncol] = hnew;
    }

    if (t + 1 < S_DIM) {
      // ---------- overlap: stage x_{t+1} and compute gi_{t+1} BEFORE the barrier ----------
      // (pure function of x; hides behind store-drain + other blocks' tail work)
      const float* xp = x_stage + (size_t)(t + 1) * H_DIM;
      *(float4*)&sx[sr][sc]     = ((const float4*)xp)[0];
      *(float4*)&sx[sr][sc + 4] = ((const float4*)xp)[1];
      if (t + 2 < S_DIM) __builtin_prefetch(xp + H_DIM, 0, 0);
      __syncthreads();

      a_ir = vzero; a_iz = vzero; a_in = vzero;
      #pragma unroll 4
      for (int kk = 0; kk < H_DIM; kk += 4) {
        const int kb = kk + kofs;
        v2f ax  = *(const v2f*)&sx[ln][kb];
        v2f bri = *(const v2f*)(wr_i + kb);
        v2f bzi = *(const v2f*)(wz_i + kb);
        v2f bni = *(const v2f*)(wn_i + kb);
        a_ir = __builtin_amdgcn_wmma_f32_16x16x4_f32(false, ax, false, bri, (short)0, a_ir, false, false);
        a_iz = __builtin_amdgcn_wmma_f32_16x16x4_f32(false, ax, false, bzi, (short)0, a_iz, false, false);
        a_in = __builtin_amdgcn_wmma_f32_16x16x4_f32(false, ax, false, bni, (short)0, a_in, false, false);
      }

      // ---------- grid-wide release/acquire barrier (step fence) ----------
      __threadfence();            // drain h_new stores, make them agent-visible
      __syncthreads();
      if (tid == 0) {
        __hip_atomic_fetch_add(bar, 1u, __ATOMIC_RELEASE, __HIP_MEMORY_SCOPE_AGENT);
        const unsigned int target = (unsigned int)N_BLOCKS * (unsigned int)(t + 1);
        while (__hip_atomic_load(bar, __ATOMIC_ACQUIRE, __HIP_MEMORY_SCOPE_AGENT) < target) {
          __builtin_amdgcn_s_sleep(1);   // short sleep: low-latency re-poll
        }
      }
      __syncthreads();
      __builtin_amdgcn_fence(__ATOMIC_ACQUIRE, "agent");  // invalidate L0 before next gather
    }
  }
}

extern "C" void kernel_launch(void* const* d_in, const int* in_sizes, int n_in,
                              void* d_out, int out_size, void* d_ws, size_t ws_size,
                              hipStream_t stream) {
  const float* x    = (const float*)d_in[0];   // [B,S,H]
  const float* w_ih = (const float*)d_in[1];   // [3H,H]
  const float* w_hh = (const float*)d_in[2];   // [3H,H]
  const float* b_ih = (const float*)d_in[3];   // [3H]
  const float* b_hh = (const float*)d_in[4];   // [3H]
  const int*   idx  = (const int*)d_in[5];     // [B,S]
  float* out = (float*)d_out;                  // [B,S,H] = recurrence buffer
  unsigned int* bar = (unsigned int*)d_ws;

  gru_bar_init<<<1, 64, 0, stream>>>(bar);
  dilated_gru_wmma<<<dim3(N_BLOCKS), dim3(256), 0, stream>>>(
      x, w_ih, w_hh, b_ih, b_hh, idx, out, bar);
}